// SlidingWindowMatrixMult3D2_73452530696596
// MI455X (gfx1250) — compile-verified
//
#include <hip/hip_runtime.h>

#define AS1 __attribute__((address_space(1)))
#define AS3 __attribute__((address_space(3)))

typedef __attribute__((ext_vector_type(2))) float v2f;
typedef __attribute__((ext_vector_type(4))) float v4f;
typedef __attribute__((ext_vector_type(8))) float v8f;
typedef __attribute__((ext_vector_type(4))) int   v4i;

#ifndef __has_builtin
#define __has_builtin(x) 0
#endif

static constexpr int Bc   = 4;
static constexpr int Tc   = 1024;
static constexpr int Cc   = 128;
static constexpr int TY   = 2 * Tc - 1;              // 2047 Y rows
static constexpr int KC   = 32;                      // u-chunk staged in LDS
static constexpr int NCH  = (Tc + 15 + KC - 1) / KC; // 33 chunks cover u in [t0, t0+1039)
static constexpr int LDSW = Cc + 4;                  // padded LDS row stride (floats)

// Stage Y[u0 : u0+32, 0:128] for batch b into ybuf via async-to-LDS DMA.
__device__ __forceinline__ void stage_y_chunk(const float* __restrict__ Y,
                                              float (*ybuf)[LDSW],
                                              int b, int u0, int tid) {
#pragma unroll
  for (int jj = 0; jj < 4; ++jj) {
    const int idx = jj * 256 + tid;        // 1024 16-byte transfers total
    const int r   = idx >> 5;              // 0..31 (chunk row)
    const int cg  = (idx & 31) << 2;       // column (floats), multiple of 4
    int grow = u0 + r;
    if (grow > TY - 1) grow = TY - 1;      // clamp (masked on the A side)
    const float* gp = Y + ((size_t)b * TY + grow) * Cc + cg;
    float*       lp = &ybuf[r][cg];
#if __has_builtin(__builtin_amdgcn_global_load_async_to_lds_b128)
    __builtin_amdgcn_global_load_async_to_lds_b128((AS1 v4i*)gp,
                                                   (AS3 v4i*)lp, 0, 0);
#else
    *(v4f*)lp = *(const v4f*)gp;           // fallback: through VGPRs
#endif
  }
}

__device__ __forceinline__ void wait_async_all() {
#if __has_builtin(__builtin_amdgcn_s_wait_asynccnt)
  __builtin_amdgcn_s_wait_asynccnt(0);
#else
  asm volatile("s_wait_asynccnt 0" ::: "memory");
#endif
}

// Branchless zeroing: keep v iff m is all-ones.
__device__ __forceinline__ float and_mask(float v, int m) {
  return __int_as_float(__float_as_int(v) & m);
}

// One 32-wide K-chunk: 8 x (16x16x4) f32 WMMAs.
// EDGE=true applies banded-edge masking (needed only for first/last chunk).
template <bool EDGE>
__device__ __forceinline__ void compute_chunk(v8f& acc,
                                              const float* __restrict__ Xrow,
                                              const float (*ybuf)[LDSW],
                                              int xbase, int ln, int h, int n0) {
  const int xoff = 2 * h - ln;                 // per-lane A column offset
  const float* ap0 = Xrow + (xbase + xoff);    // one address setup per chunk
#pragma unroll
  for (int j = 0; j < 8; ++j) {
    const int kb = 4 * j + 2 * h;              // k_local of this lane's A pair
    v2f a;
    if (EDGE) {
      const int x0 = xbase + 4 * j + xoff;     // x = u - (t0+ln)
      const int x1 = x0 + 1;
      int c0 = x0 < 0 ? 0 : (x0 > Tc - 1 ? Tc - 1 : x0);
      int c1 = x1 < 0 ? 0 : (x1 > Tc - 1 ? Tc - 1 : x1);
      const int m0 = ((unsigned)x0 < (unsigned)Tc) ? -1 : 0;
      const int m1 = ((unsigned)x1 < (unsigned)Tc) ? -1 : 0;
      a[0] = and_mask(Xrow[c0], m0);           // unconditional load + AND-mask
      a[1] = and_mask(Xrow[c1], m1);
    } else {                                   // interior: always in-band
      a[0] = ap0[4 * j + 0];                   // const offsets -> b64 + ioffset
      a[1] = ap0[4 * j + 1];
    }
    v2f bb;                                    // B from LDS: rows kb, kb+1
    bb[0] = ybuf[kb + 0][n0 + ln];
    bb[1] = ybuf[kb + 1][n0 + ln];
    acc = __builtin_amdgcn_wmma_f32_16x16x4_f32(
        /*neg_a=*/false, a, /*neg_b=*/false, bb,
        /*c_mod=*/(short)0, acc, /*reuse_a=*/false, /*reuse_b=*/false);
  }
}

__global__ __launch_bounds__(256)
void sliding_window_wmma_f32_kernel(const float* __restrict__ X,
                                    const float* __restrict__ Y,
                                    float* __restrict__ out) {
  __shared__ float Ybuf[2][KC][LDSW];      // double-buffered Y stage (~33 KB)

  const int tid  = threadIdx.x;
  const int wave = tid >> 5;               // 8 waves; wave owns N-tile 16*wave
  const int lane = tid & 31;
  const int ln   = lane & 15;              // M index for A, N index for B/C/D
  const int h    = lane >> 4;              // lane half selects K pair / M half
  const int b    = blockIdx.y;
  const int t0   = blockIdx.x << 4;        // 16-row output tile
  const int n0   = wave << 4;

  // A source row: X[b, t0+ln, :]
  const float* Xrow = X + ((size_t)b * Tc + (t0 + ln)) * Tc;

  v8f acc = {0.f, 0.f, 0.f, 0.f, 0.f, 0.f, 0.f, 0.f};

  // Prologue: stage chunk 0, then compute it (edge-masked) while chunk 1 streams.
  stage_y_chunk(Y, Ybuf[0], b, t0, tid);
  wait_async_all();
  __syncthreads();

  stage_y_chunk(Y, Ybuf[1], b, t0 + KC, tid);
  compute_chunk<true>(acc, Xrow, Ybuf[0], 0, ln, h, n0);
  wait_async_all();
  __syncthreads();

  // Interior chunks: no band masking, tight loop.
  for (int ch = 1; ch < NCH - 1; ++ch) {
    const int cur = ch & 1;
    stage_y_chunk(Y, Ybuf[cur ^ 1], b, t0 + (ch + 1) * KC, tid);
    compute_chunk<false>(acc, Xrow, Ybuf[cur], ch * KC, ln, h, n0);
    wait_async_all();
    __syncthreads();
  }

  // Final chunk (edge-masked); its data is already resident.
  compute_chunk<true>(acc, Xrow, Ybuf[(NCH - 1) & 1], (NCH - 1) * KC, ln, h, n0);

  // C/D layout: VGPR v -> M = v + 8*h, N = ln
#pragma unroll
  for (int v = 0; v < 8; ++v) {
    const int row = t0 + v + 8 * h;
    out[((size_t)b * Tc + row) * Cc + n0 + ln] = acc[v];
  }
}

extern "C" void kernel_launch(void* const* d_in, const int* in_sizes, int n_in,
                              void* d_out, int out_size, void* d_ws, size_t ws_size,
                              hipStream_t stream) {
  const float* X = (const float*)d_in[0];   // (4, 1024, 1024) f32
  const float* Y = (const float*)d_in[1];   // (4, 2047, 128)  f32
  float* out = (float*)d_out;               // (4, 1024, 128)  f32
  dim3 grid(Tc / 16, Bc);                   // 64 row-tiles x 4 batches
  sliding_window_wmma_f32_kernel<<<grid, 256, 0, stream>>>(X, Y, out);
}